// MaskedNodePredictorWithEncoder_17291538334013
// MI455X (gfx1250) — compile-verified
//
#include <hip/hip_runtime.h>
#include <hip/hip_bf16.h>

typedef float v2f __attribute__((ext_vector_type(2)));
typedef float v8f __attribute__((ext_vector_type(8)));

#define GN     50000      // nodes
#define GE     500000     // edges
#define GEF    (GE + GN)  // edges + self loops
#define HC     256        // heads*channels
#define CC     64         // channels per head
#define NMASK  7500
#define FIN    128

__device__ __forceinline__ float gelu_f(float x) {
    return 0.5f * x * (1.0f + erff(x * 0.70710678118654752f));
}
__device__ __forceinline__ unsigned flipf(float f) {
    unsigned u = __float_as_uint(f);
    return (u & 0x80000000u) ? ~u : (u | 0x80000000u);
}
__device__ __forceinline__ float unflipf(unsigned u) {
    return (u & 0x80000000u) ? __uint_as_float(u & 0x7FFFFFFFu) : __uint_as_float(~u);
}

// ---------------------------------------------------------------------------
// fp32 WMMA GEMM: C[M,N] = act(A[M,K] @ W[K,N] + bias)
// Each wave computes a 16x64 strip: 4 independent accumulators (4 WMMA dep
// chains interleave -> hides XDL latency), A-fragment loaded once per k-step
// and reused by all 4 V_WMMA_F32_16X16X4_F32.
// A-frag (16x4 f32): lanes 0-15 rows M0..15 {K0,K1}; lanes 16-31 {K2,K3}.
// B-frag (4x16 f32): v0 = K0(lanes0-15)/K2(lanes16-31), v1 = K1/K3.
// C/D: VGPR r -> M=r (lanes 0-15) / M=r+8 (lanes 16-31), N=lane%16.
// M-guard: clamp load row (EXEC stays all ones for WMMA), predicate stores.
// Requires N % 64 == 0 and K % 4 == 0 (true for all GEMMs here).
// ---------------------------------------------------------------------------
__global__ void gemm_wmma_f32(const float* __restrict__ A, const float* __restrict__ W,
                              const float* __restrict__ bias, float* __restrict__ C,
                              int M, int K, int N, int act)
{
    const int lane   = threadIdx.x & 31;
    const int wave   = threadIdx.x >> 5;
    const int strips = N >> 6;                       // 16x64 strips per row
    const int tilesM = (M + 15) >> 4;
    int tile = blockIdx.x * (blockDim.x >> 5) + wave;
    if (tile >= tilesM * strips) return;             // wave-uniform
    const int tm   = tile / strips;
    const int tn   = tile - tm * strips;             // strip index
    const int half = lane >> 4;
    const int l16  = lane & 15;
    int rowA = tm * 16 + l16;
    if (rowA > M - 1) rowA = M - 1;                  // clamp, keep EXEC full
    const int col0 = tn * 64 + l16;                  // first of 4 tile columns
    const float* __restrict__ arow = A + (size_t)rowA * K;

    v8f acc0 = {0.f,0.f,0.f,0.f,0.f,0.f,0.f,0.f};
    v8f acc1 = acc0, acc2 = acc0, acc3 = acc0;

    for (int k0 = 0; k0 < K; k0 += 4) {
        v2f a;
        a[0] = arow[k0 + 2 * half + 0];
        a[1] = arow[k0 + 2 * half + 1];
        const float* __restrict__ w0 = W + (size_t)(k0 + 2 * half + 0) * N + col0;
        const float* __restrict__ w1 = W + (size_t)(k0 + 2 * half + 1) * N + col0;
        v2f b0, b1, b2, b3;
        b0[0] = w0[0];  b0[1] = w1[0];
        b1[0] = w0[16]; b1[1] = w1[16];
        b2[0] = w0[32]; b2[1] = w1[32];
        b3[0] = w0[48]; b3[1] = w1[48];
        acc0 = __builtin_amdgcn_wmma_f32_16x16x4_f32(false, a, false, b0, (short)0, acc0, false, false);
        acc1 = __builtin_amdgcn_wmma_f32_16x16x4_f32(false, a, false, b1, (short)0, acc1, false, false);
        acc2 = __builtin_amdgcn_wmma_f32_16x16x4_f32(false, a, false, b2, (short)0, acc2, false, false);
        acc3 = __builtin_amdgcn_wmma_f32_16x16x4_f32(false, a, false, b3, (short)0, acc3, false, false);
    }

#pragma unroll
    for (int j = 0; j < 4; ++j) {
        const v8f& acc = (j == 0) ? acc0 : (j == 1) ? acc1 : (j == 2) ? acc2 : acc3;
        const int col = col0 + j * 16;
        const float bv = bias ? bias[col] : 0.f;
#pragma unroll
        for (int r = 0; r < 8; ++r) {
            int row = tm * 16 + r + 8 * half;
            if (row < M) {
                float v = acc[r] + bv;
                if (act == 1) v = gelu_f(v);
                C[(size_t)row * N + col] = v;
            }
        }
    }
}

// ---------------------------------------------------------------------------
// Self-loop attr: deg[d] += 1 ; asum[d,:] += edge_attr[e,:]
// ---------------------------------------------------------------------------
__global__ void k_deg_attr(const float* __restrict__ ea, const int* __restrict__ dst,
                           float* __restrict__ deg, float* __restrict__ asum)
{
    int e = blockIdx.x * blockDim.x + threadIdx.x;
    if (e >= GE) return;
    int d = dst[e];
    atomicAdd(&deg[d], 1.f);
#pragma unroll
    for (int k = 0; k < 16; ++k)
        atomicAdd(&asum[(size_t)d * 16 + k], ea[(size_t)e * 16 + k]);
}

__global__ void k_loop_div(float* __restrict__ loop, const float* __restrict__ deg)
{
    int i = blockIdx.x * blockDim.x + threadIdx.x;
    if (i >= GN * 16) return;
    float dg = deg[i >> 4];
    loop[i] /= (dg < 1.f ? 1.f : dg);
}

// ---------------------------------------------------------------------------
// Logits: per (edge-with-selfloops, head). Fuses ea@W_e (W_e + att in LDS).
// logit = sum_c lrelu(xl[src]+xr[dst]+ef) * att ; atomicMax(m) via uint flip.
// ---------------------------------------------------------------------------
__global__ void k_edge_logits(const float* __restrict__ XL, const float* __restrict__ XR,
                              const float* __restrict__ edge_attr,
                              const float* __restrict__ loop_attr,
                              const int* __restrict__ src, const int* __restrict__ dst,
                              const float* __restrict__ W_e, const float* __restrict__ att,
                              float* __restrict__ logits, unsigned* __restrict__ mflip)
{
    __shared__ float sWe[16 * HC];
    __shared__ float sAtt[HC];
    for (int i = threadIdx.x; i < 16 * HC; i += blockDim.x) sWe[i] = W_e[i];
    for (int i = threadIdx.x; i < HC; i += blockDim.x) sAtt[i] = att[i];
    __syncthreads();

    int gid = blockIdx.x * blockDim.x + threadIdx.x;
    if (gid >= GEF * 4) return;
    int ef = gid >> 2, h = gid & 3;
    int s, d;
    const float* ea;
    if (ef < GE) { s = src[ef]; d = dst[ef]; ea = edge_attr + (size_t)ef * 16; }
    else         { s = d = ef - GE;          ea = loop_attr + (size_t)(ef - GE) * 16; }

    float eav[16];
#pragma unroll
    for (int k = 0; k < 16; ++k) eav[k] = ea[k];

    const float* __restrict__ xl = XL + (size_t)s * HC + h * CC;
    const float* __restrict__ xr = XR + (size_t)d * HC + h * CC;
    float acc = 0.f;
#pragma unroll 4
    for (int c = 0; c < CC; ++c) {
        float efc = 0.f;
#pragma unroll
        for (int k = 0; k < 16; ++k) efc = fmaf(eav[k], sWe[k * HC + h * CC + c], efc);
        float z = xl[c] + xr[c] + efc;
        z = z > 0.f ? z : 0.2f * z;          // leaky_relu(0.2)
        acc = fmaf(z, sAtt[h * CC + c], acc);
    }
    logits[gid] = acc;
    atomicMax(&mflip[(size_t)d * 4 + h], flipf(acc));
}

// exp(logit - m[dst]) in-place; den[dst] += ex
__global__ void k_edge_expsum(float* __restrict__ logits, const unsigned* __restrict__ mflip,
                              float* __restrict__ den, const int* __restrict__ dst)
{
    int gid = blockIdx.x * blockDim.x + threadIdx.x;
    if (gid >= GEF * 4) return;
    int ef = gid >> 2, h = gid & 3;
    int d = (ef < GE) ? dst[ef] : ef - GE;
    float m = unflipf(mflip[(size_t)d * 4 + h]);
    float e = __expf(logits[gid] - m);
    logits[gid] = e;
    atomicAdd(&den[(size_t)d * 4 + h], e);
}

// scat[dst, :] += XL[src, :] * alpha   (float4 per thread, 4 atomics)
__global__ void k_edge_scatter(const float* __restrict__ XL, const float* __restrict__ ex,
                               const float* __restrict__ den,
                               const int* __restrict__ src, const int* __restrict__ dst,
                               float* __restrict__ scat)
{
    int gid = blockIdx.x * blockDim.x + threadIdx.x;
    if (gid >= GEF * (HC / 4)) return;
    int ef = gid >> 6;
    int q  = gid & 63;                 // float4 index; channel = q*4
    int h  = q >> 4;                   // (q*4)/64
    int s, d;
    if (ef < GE) { s = src[ef]; d = dst[ef]; } else { s = d = ef - GE; }
    float a = ex[(size_t)ef * 4 + h] / (den[(size_t)d * 4 + h] + 1e-16f);
    float4 v = *((const float4*)(XL + (size_t)s * HC) + q);
    float* o = scat + (size_t)d * HC + q * 4;
    atomicAdd(o + 0, v.x * a);
    atomicAdd(o + 1, v.y * a);
    atomicAdd(o + 2, v.z * a);
    atomicAdd(o + 3, v.w * a);
}

// h1 = gelu(LN(scat + xres + bias))   (256 ch/node, one block per node)
__global__ void k_post1(const float* __restrict__ scat, const float* __restrict__ xres,
                        const float* __restrict__ bias, const float* __restrict__ g,
                        const float* __restrict__ be, float* __restrict__ h1)
{
    int n = blockIdx.x, c = threadIdx.x;           // 256 threads
    __shared__ float red[HC];
    float v = scat[(size_t)n * HC + c] + xres[(size_t)n * HC + c] + bias[c];
    red[c] = v; __syncthreads();
    for (int s = HC / 2; s > 0; s >>= 1) { if (c < s) red[c] += red[c + s]; __syncthreads(); }
    float mu = red[0] * (1.f / HC); __syncthreads();
    float dv = v - mu;
    red[c] = dv * dv; __syncthreads();
    for (int s = HC / 2; s > 0; s >>= 1) { if (c < s) red[c] += red[c + s]; __syncthreads(); }
    float var = red[0] * (1.f / HC);
    float y = dv * rsqrtf(var + 1e-5f) * g[c] + be[c];
    h1[(size_t)n * HC + c] = gelu_f(y);
}

// out = LN(m2 + t)   (64 ch/node)
__global__ void k_post2(const float* __restrict__ m2, const float* __restrict__ t,
                        const float* __restrict__ g, const float* __restrict__ be,
                        float* __restrict__ out)
{
    int n = blockIdx.x, c = threadIdx.x;           // 64 threads
    __shared__ float red[CC];
    float v = m2[(size_t)n * CC + c] + t[(size_t)n * CC + c];
    red[c] = v; __syncthreads();
    for (int s = CC / 2; s > 0; s >>= 1) { if (c < s) red[c] += red[c + s]; __syncthreads(); }
    float mu = red[0] * (1.f / CC); __syncthreads();
    float dv = v - mu;
    red[c] = dv * dv; __syncthreads();
    for (int s = CC / 2; s > 0; s >>= 1) { if (c < s) red[c] += red[c + s]; __syncthreads(); }
    float var = red[0] * (1.f / CC);
    out[(size_t)n * CC + c] = dv * rsqrtf(var + 1e-5f) * g[c] + be[c];
}

__global__ void k_gather(const float* __restrict__ h, const int* __restrict__ mask,
                         float* __restrict__ mbuf)
{
    int gid = blockIdx.x * blockDim.x + threadIdx.x;
    if (gid >= NMASK * CC) return;
    int i = gid >> 6, c = gid & 63;
    mbuf[gid] = h[(size_t)mask[i] * CC + c];
}

// ---------------------------------------------------------------------------
extern "C" void kernel_launch(void* const* d_in, const int* in_sizes, int n_in,
                              void* d_out, int out_size, void* d_ws, size_t ws_size,
                              hipStream_t stream)
{
    (void)in_sizes; (void)n_in; (void)out_size; (void)ws_size;
    // Input layout (insertion order of setup_inputs + nested params dicts):
    // 0:x 1:edge_attr 2..19:layer0 20..37:layer1 38..41:pred 42:edge_index 43:mask
    const float* x         = (const float*)d_in[0];
    const float* edge_attr = (const float*)d_in[1];
    const int*   edge_idx  = (const int*)d_in[42];
    const int*   src       = edge_idx;
    const int*   dst       = edge_idx + GE;
    const int*   mask      = (const int*)d_in[43];

    // workspace arena (floats)
    float* ws = (float*)d_ws;
    size_t off = 0;
    auto alloc = [&](size_t n) { size_t o = off; off += (n + 63) & ~(size_t)63; return ws + o; };
    float*    f_deg  = alloc(GN);
    float*    f_loop = alloc((size_t)GN * 16);
    float*    f_XL   = alloc((size_t)GN * HC);
    float*    f_XR   = alloc((size_t)GN * HC);
    float*    f_XRES = alloc((size_t)GN * HC);
    float*    f_log  = alloc((size_t)GEF * 4);
    unsigned* f_m    = (unsigned*)alloc((size_t)GN * 4);
    float*    f_den  = alloc((size_t)GN * 4);
    float*    f_scat = alloc((size_t)GN * HC);
    float*    f_h1   = alloc((size_t)GN * HC);
    float*    f_t    = alloc((size_t)GN * CC);
    float*    f_m1   = alloc((size_t)GN * CC);
    float*    f_m2   = alloc((size_t)GN * CC);
    float*    f_hout = alloc((size_t)GN * CC);
    float*    f_mbuf = alloc((size_t)NMASK * CC);
    float*    f_p1   = alloc((size_t)NMASK * CC);

    auto gemm = [&](const float* A, const float* W, const float* b, float* C,
                    int M, int K, int N, int act) {
        int tiles  = ((M + 15) / 16) * (N / 64);   // 16x64 strips
        int blocks = (tiles + 7) / 8;              // 8 waves per block
        hipLaunchKernelGGL(gemm_wmma_f32, dim3(blocks), dim3(256), 0, stream,
                           A, W, b, C, M, K, N, act);
    };

    // ---- self-loop edge attr (mean of incoming), once ----
    hipMemsetAsync(f_deg, 0, (size_t)GN * 4, stream);
    hipMemsetAsync(f_loop, 0, (size_t)GN * 16 * 4, stream);
    hipLaunchKernelGGL(k_deg_attr, dim3((GE + 255) / 256), dim3(256), 0, stream,
                       edge_attr, dst, f_deg, f_loop);
    hipLaunchKernelGGL(k_loop_div, dim3((GN * 16 + 255) / 256), dim3(256), 0, stream,
                       f_loop, f_deg);

    const float* XP = x;
    for (int L = 0; L < 2; ++L) {
        const int base = (L == 0) ? 2 : 20;
        const int DL   = (L == 0) ? FIN : CC;
        const float* W_l   = (const float*)d_in[base + 0];
        const float* b_l   = (const float*)d_in[base + 1];
        const float* W_r   = (const float*)d_in[base + 2];
        const float* b_r   = (const float*)d_in[base + 3];
        const float* W_e   = (const float*)d_in[base + 4];
        const float* att   = (const float*)d_in[base + 5];
        const float* W_res = (const float*)d_in[base + 6];
        const float* bias  = (const float*)d_in[base + 7];
        const float* g1    = (const float*)d_in[base + 8];
        const float* be1   = (const float*)d_in[base + 9];
        const float* W_h   = (const float*)d_in[base + 10];
        const float* b_h   = (const float*)d_in[base + 11];
        const float* W_m1  = (const float*)d_in[base + 12];
        const float* b_m1  = (const float*)d_in[base + 13];
        const float* W_m2  = (const float*)d_in[base + 14];
        const float* b_m2  = (const float*)d_in[base + 15];
        const float* g2    = (const float*)d_in[base + 16];
        const float* be2   = (const float*)d_in[base + 17];

        gemm(XP, W_l,   b_l,     f_XL,   GN, DL, HC, 0);
        gemm(XP, W_r,   b_r,     f_XR,   GN, DL, HC, 0);
        gemm(XP, W_res, nullptr, f_XRES, GN, DL, HC, 0);

        hipMemsetAsync(f_m,   0, (size_t)GN * 4 * 4, stream);   // flip(-inf) id
        hipMemsetAsync(f_den, 0, (size_t)GN * 4 * 4, stream);
        hipMemsetAsync(f_scat, 0, (size_t)GN * HC * 4, stream);

        hipLaunchKernelGGL(k_edge_logits, dim3((GEF * 4 + 255) / 256), dim3(256), 0, stream,
                           f_XL, f_XR, edge_attr, f_loop, src, dst, W_e, att, f_log, f_m);
        hipLaunchKernelGGL(k_edge_expsum, dim3((GEF * 4 + 255) / 256), dim3(256), 0, stream,
                           f_log, f_m, f_den, dst);
        hipLaunchKernelGGL(k_edge_scatter, dim3((GEF * 64 + 255) / 256), dim3(256), 0, stream,
                           f_XL, f_log, f_den, src, dst, f_scat);

        hipLaunchKernelGGL(k_post1, dim3(GN), dim3(HC), 0, stream,
                           f_scat, f_XRES, bias, g1, be1, f_h1);

        gemm(f_h1, W_h,  b_h,  f_t,  GN, HC, CC, 0);
        gemm(f_t,  W_m1, b_m1, f_m1, GN, CC, CC, 1);   // fused gelu
        gemm(f_m1, W_m2, b_m2, f_m2, GN, CC, CC, 0);

        hipLaunchKernelGGL(k_post2, dim3(GN), dim3(CC), 0, stream,
                           f_m2, f_t, g2, be2, f_hout);
        XP = f_hout;
    }

    // ---- predictor on masked nodes ----
    const float* W_p1 = (const float*)d_in[38];
    const float* b_p1 = (const float*)d_in[39];
    const float* W_p2 = (const float*)d_in[40];
    const float* b_p2 = (const float*)d_in[41];

    hipLaunchKernelGGL(k_gather, dim3((NMASK * CC + 255) / 256), dim3(256), 0, stream,
                       f_hout, mask, f_mbuf);
    gemm(f_mbuf, W_p1, b_p1, f_p1, NMASK, CC, CC, 1);
    gemm(f_p1, W_p2, b_p2, (float*)d_out, NMASK, CC, FIN, 0);
}